// RansEmbedding_44598940402133
// MI455X (gfx1250) — compile-verified
//
#include <hip/hip_runtime.h>
#include <hip/hip_bf16.h>
#include <stdint.h>

// ---------------- problem constants (match reference) ----------------
#define NUM_EMB   65536
#define DIM       256
#define TILE_K    1024
#define TILE_N    32
#define TILE_ELEMS (TILE_K * TILE_N)        // 32768
#define COL_TILES 8
#define NUM_TILES 512
#define STREAM_LEN (2 * TILE_ELEMS + 8)     // 65544 bytes per tile stream
#define PROB_BITS 12
#define MSLOTS    4096
#define RANS_LO   (1u << 23)

// ---------------- implementation constants ----------------
#define PACKED_STRIDE 65664                 // STREAM_LEN rounded up to 64B + slack
#define BUFSZ     256                       // per-chain LDS circular byte buffer
#define T_SLOTS   2                         // interleaved rANS chains per lane
#define BATCH     16                        // symbols per chain between refill checks
#define DEC_BLOCKS (NUM_TILES / (32 * T_SLOTS))   // 8 single-wave workgroups

// gfx1250 async global->LDS path (guarded; pure-C fallback otherwise)
#if defined(__has_builtin)
#if __has_builtin(__builtin_amdgcn_global_load_async_to_lds_b128) && \
    __has_builtin(__builtin_amdgcn_s_wait_asynccnt)
#define USE_ASYNC_LDS 1
#endif
#endif

#if defined(USE_ASYNC_LDS)
// builtin signature (confirmed by hipcc diagnostic): (v4i AS1*, v4i AS3*, Iint, Iint)
typedef int v4i_t __attribute__((ext_vector_type(4)));
typedef __attribute__((address_space(1))) v4i_t* gv4_t;   // global
typedef __attribute__((address_space(3))) v4i_t* lv4_t;   // LDS
#endif

// =====================================================================
// Kernel 1: pack int32-per-byte stream into contiguous bytes per tile
// (coalesced, pure bandwidth; enables b128 refills in the decode loop)
// =====================================================================
__global__ __launch_bounds__(256) void k_pack(const int* __restrict__ s,
                                              unsigned char* __restrict__ p) {
  const int tile = blockIdx.y;
  const int j = blockIdx.x * 256 + threadIdx.x;       // dword-group within tile
  if (j >= STREAM_LEN / 4) return;
  const int4 v = ((const int4*)(s + (size_t)tile * STREAM_LEN))[j];
  const unsigned int b =
      ((unsigned)v.x & 0xFFu) | (((unsigned)v.y & 0xFFu) << 8) |
      (((unsigned)v.z & 0xFFu) << 16) | (((unsigned)v.w & 0xFFu) << 24);
  *(unsigned int*)(p + (size_t)tile * PACKED_STRIDE + 4 * (size_t)j) = b;
}

// ---- 64-byte refill of one chain's LDS circular buffer ----
__device__ __forceinline__ void refill64(const unsigned char* __restrict__ packed,
                                         unsigned int goff, unsigned int fillv,
                                         unsigned char* __restrict__ sbuf,
                                         unsigned int loff) {
  const unsigned char* g = packed + goff + fillv;
  unsigned char* l = sbuf + loff + (fillv & (BUFSZ - 1));
#if defined(USE_ASYNC_LDS)
  gv4_t gp = (gv4_t)g;
  lv4_t lp = (lv4_t)l;
#pragma unroll
  for (int q = 0; q < 4; ++q)
    __builtin_amdgcn_global_load_async_to_lds_b128(gp + q, lp + q, 0, 0);
#else
#pragma unroll
  for (int q = 0; q < 4; ++q)
    ((uint4*)l)[q] = ((const uint4*)g)[q];
#endif
}

__device__ __forceinline__ void async_fence() {
#if defined(USE_ASYNC_LDS)
  __builtin_amdgcn_s_wait_asynccnt(0);
  asm volatile("" ::: "memory");
#endif
}

// =====================================================================
// Kernel 2: rANS decode. 8 single-wave workgroups, 2 interleaved chains
// per lane = 512 decoders. Iteration time sits at the per-symbol chain
// latency floor (one ds_load_b32 of the fused table + ~8 VALU); the two
// chains hide each other's LDS latency. Stream bytes arrive via the
// gfx1250 async global->LDS engine (ASYNCcnt / s_wait_asynccnt).
// =====================================================================
__global__ __launch_bounds__(32) void k_decode(
    const unsigned char* __restrict__ packed,
    const unsigned int* __restrict__ states,
    const int* __restrict__ tables,
    const int* __restrict__ slot_map,
    unsigned char* __restrict__ hi) {
  __shared__ unsigned int s_tab[MSLOTS];                   // 16 KB fused table
  __shared__ unsigned char s_buf[32 * T_SLOTS * BUFSZ];    // 16 KB stream buffers

  const int lane = threadIdx.x;
  // Build fused per-slot table: (sym<<24)|(freq<<12)|cum  (freq,cum < 4096)
  for (int i = lane; i < MSLOTS; i += 32) {
    const int sym = slot_map[i];
    const unsigned int t = (unsigned int)tables[sym];
    s_tab[i] = ((unsigned int)sym << 24) | ((t >> 16) << 12) | (t & 0xFFFu);
  }
  __syncthreads();

  unsigned int st[T_SLOTS], pos[T_SLOTS], fill[T_SLOTS], outw[T_SLOTS];
  unsigned int hib[T_SLOTS], gsb[T_SLOTS], lb[T_SLOTS];
#pragma unroll
  for (int j = 0; j < T_SLOTS; ++j) {
    const int tile = blockIdx.x * (32 * T_SLOTS) + j * 32 + lane;
    st[j] = states[tile];
    pos[j] = 0u; fill[j] = 0u; outw[j] = 0u;
    hib[j] = (unsigned)(tile >> 3) * (unsigned)(TILE_K * DIM) +
             (unsigned)(tile & (COL_TILES - 1)) * TILE_N;
    gsb[j] = (unsigned)tile * (unsigned)PACKED_STRIDE;
    lb[j]  = (unsigned)(lane * T_SLOTS + j) * BUFSZ;
  }
  // Prefill 128 B per chain.
  // Invariant: avail >= 96 at every batch start; max consumption/batch = 32 B;
  // refill (+64 B) when avail < 128 keeps the 256 B ring from overwriting
  // unread bytes (write window avail+64 <= 191 < 256).
#pragma unroll
  for (int r = 0; r < 2; ++r)
#pragma unroll
    for (int j = 0; j < T_SLOTS; ++j) {
      refill64(packed, gsb[j], fill[j], s_buf, lb[j]);
      fill[j] += 64u;
    }
  async_fence();

  for (int kk = 0; kk < TILE_ELEMS; kk += BATCH) {
#pragma unroll
    for (int j = 0; j < T_SLOTS; ++j) {
      if (fill[j] - pos[j] < 128u && fill[j] < (unsigned)PACKED_STRIDE) {
        refill64(packed, gsb[j], fill[j], s_buf, lb[j]);
        fill[j] += 64u;
      }
    }
    async_fence();

    const unsigned int rowoff = ((unsigned)kk >> 5) << 8;  // (k/32)*256
    const unsigned int coloff = (unsigned)kk & 31u;        // kk%32 in {0,16}
#pragma unroll
    for (int u = 0; u < BATCH; ++u) {
#pragma unroll
      for (int j = 0; j < T_SLOTS; ++j) {
        unsigned int state = st[j];
        const unsigned int slot = state & (MSLOTS - 1);
        const unsigned int pk = s_tab[slot];
        const unsigned int sym = pk >> 24;
        state = ((pk >> 12) & 0xFFFu) * (state >> PROB_BITS) + slot - (pk & 0xFFFu);
        // branchless renorm, bit-exact vs reference (2 speculative byte reads)
        const unsigned int b0 = s_buf[lb[j] + (pos[j] & (BUFSZ - 1))];
        const unsigned int b1 = s_buf[lb[j] + ((pos[j] + 1u) & (BUFSZ - 1))];
        const bool t0 = state < RANS_LO;
        state = t0 ? ((state << 8) | b0) : state;
        const bool t1 = state < RANS_LO;             // false automatically if !t0
        state = t1 ? ((state << 8) | b1) : state;
        pos[j] += (unsigned)t0 + (unsigned)t1;
        st[j] = state;
        outw[j] |= sym << ((u & 3) * 8);
        if ((u & 3) == 3) {                          // one b32 store per 4 syms
          *(unsigned int*)(hi + (size_t)(hib[j] + rowoff + coloff + (unsigned)u - 3u)) = outw[j];
          outw[j] = 0u;
        }
      }
    }
  }
}

// =====================================================================
// Kernel 3: assemble bf16 weight table: bits = (hi<<8) | man  (32 MB)
// =====================================================================
__global__ __launch_bounds__(256) void k_assemble(const unsigned char* __restrict__ hi,
                                                  const int* __restrict__ man,
                                                  unsigned int* __restrict__ w) {
  const int i = blockIdx.x * 256 + threadIdx.x;   // 4 elements per thread
  const unsigned int h = ((const unsigned int*)hi)[i];
  const int4 m = ((const int4*)man)[i];
  const unsigned int w01 =
      (((h & 0xFFu) << 8) | ((unsigned)m.x & 0xFFu)) |
      ((((((h >> 8) & 0xFFu) << 8) | ((unsigned)m.y & 0xFFu))) << 16);
  const unsigned int w23 =
      ((((h >> 16) & 0xFFu) << 8) | ((unsigned)m.z & 0xFFu)) |
      ((((((h >> 24) & 0xFFu) << 8) | ((unsigned)m.w & 0xFFu))) << 16);
  ((uint2*)w)[i] = make_uint2(w01, w23);
}

// =====================================================================
// Kernel 4: fused embedding gather, bf16 -> f32 (table is L2-resident)
// =====================================================================
__global__ __launch_bounds__(256) void k_gather(const int* __restrict__ x,
                                                const unsigned short* __restrict__ w,
                                                float* __restrict__ out) {
  const int tid = threadIdx.x;
  const int row = blockIdx.x * 4 + (tid >> 6);    // 4 rows per block
  const int c = (tid & 63) << 2;                  // 4 cols per thread
  const int idx = x[row];
  const unsigned short* wr = w + ((size_t)idx << 8);
  __builtin_prefetch(wr + c, 0, 0);               // global_prefetch_b8 (gfx1250)
  const uint2 v = *(const uint2*)(wr + c);
  float4 f;
  f.x = __uint_as_float((v.x & 0xFFFFu) << 16);
  f.y = __uint_as_float(v.x & 0xFFFF0000u);
  f.z = __uint_as_float((v.y & 0xFFFFu) << 16);
  f.w = __uint_as_float(v.y & 0xFFFF0000u);
  *(float4*)(out + ((size_t)row << 8) + c) = f;
}

// =====================================================================
extern "C" void kernel_launch(void* const* d_in, const int* in_sizes, int n_in,
                              void* d_out, int out_size, void* d_ws, size_t ws_size,
                              hipStream_t stream) {
  (void)in_sizes; (void)n_in; (void)out_size; (void)ws_size;
  const int*          x       = (const int*)d_in[0];
  const int*          rstream = (const int*)d_in[1];
  const unsigned int* states  = (const unsigned int*)d_in[2];
  const int*          tables  = (const int*)d_in[3];
  const int*          slotmap = (const int*)d_in[4];
  // d_in[5] tile_offsets (uniform i*STREAM_LEN, same assumption as reference)
  // d_in[6] tile_max_lens (unused by reference)
  const int*          man     = (const int*)d_in[7];
  float*              out     = (float*)d_out;

  unsigned char* ws     = (unsigned char*)d_ws;
  unsigned char* packed = ws;                                          // 33,619,968 B
  unsigned char* hi     = packed + (size_t)NUM_TILES * PACKED_STRIDE;  // 16,777,216 B
  unsigned int*  w      = (unsigned int*)(hi + (size_t)NUM_EMB * DIM); // 33,554,432 B

  dim3 pg((STREAM_LEN / 4 + 255) / 256, NUM_TILES);
  k_pack<<<pg, 256, 0, stream>>>(rstream, packed);
  k_decode<<<DEC_BLOCKS, 32, 0, stream>>>(packed, states, tables, slotmap, hi);
  k_assemble<<<(NUM_EMB * DIM / 4) / 256, 256, 0, stream>>>(hi, man, w);
  k_gather<<<(2048 * 128) / 4, 256, 0, stream>>>(x, (const unsigned short*)w, out);
}